// patlak_model_24936580120829
// MI455X (gfx1250) — compile-verified
//
#include <hip/hip_runtime.h>
#include <hip/hip_bf16.h>
#include <stdint.h>

#define HW   65536   // 256*256
#define NT   60
#define NPIX_PER_THREAD 4

// Constants folded from the reference:
//   Ktrans = p0 / SCALE_KTRANS / 60 ; vp = p1 / SCALE_VP
//   Q  = R1*(TR/1000)*Ct = 0.0225 * Ct
//   P  = TR/(T1+eps) = 5/(T1+1e-8)
//   cf = cos(10 deg)
#define QS      0.0225f
#define CF      0.98480775301220805936f
#define INV60   (1.0f/60.0f)
#define EPSF    1e-8f

__global__ __launch_bounds__(256) void patlak_st_kernel(
    const float* __restrict__ params,  // (B,2,H,W)
    const float* __restrict__ Cp,      // (B,T)
    const float* __restrict__ S0,      // (B,1,H,W)
    const float* __restrict__ T1,      // (B,1,H,W)
    float* __restrict__ out)           // (B,T,H,W)
{
    __shared__ float  sRaw[NT];     // raw Cp[b,:]
    __shared__ float2 sCpCe[NT];    // {Cp[t], Ce[t]}  (Ce = cumsum(Cp)*DELTT)

    const int tid = blockIdx.x * 256 + threadIdx.x;   // 0..65535
    const int b   = tid >> 14;                        // uniform per block
    const int pix = (tid & 16383) << 2;               // 4 consecutive pixels

    // ---- Stage Cp[b,:] into LDS with CDNA5 async global->LDS DMA ----------
    if (threadIdx.x < NT) {
        const float* gsrc = Cp + (blockIdx.x >> 6) * NT;     // scalar (SADDR)
        unsigned vOff   = threadIdx.x * 4u;                  // per-lane byte offset
        unsigned ldsDst = (unsigned)(uintptr_t)(&sRaw[threadIdx.x]); // LDS byte addr
        asm volatile("global_load_async_to_lds_b32 %0, %1, %2"
                     :: "v"(ldsDst), "v"(vOff), "s"(gsrc) : "memory");
    }
    asm volatile("s_wait_asynccnt 0x0" ::: "memory");
    __syncthreads();

    if (threadIdx.x == 0) {
        float acc = 0.0f;
        for (int t = 0; t < NT; ++t) {
            float c = sRaw[t];
            acc += c;
            sCpCe[t] = make_float2(c, acc * 4.0f);   // DELTT = 4
        }
    }
    __syncthreads();

    // ---- Per-pixel (t-invariant) precompute, 4 pixels per thread ----------
    const float4 kt4 = *(const float4*)(params + (size_t)(2*b    ) * HW + pix);
    const float4 vp4 = *(const float4*)(params + (size_t)(2*b + 1) * HW + pix);
    const float4 s04 = *(const float4*)(S0     + (size_t)b * HW + pix);
    const float4 t14 = *(const float4*)(T1     + (size_t)b * HW + pix);

    const float kt[4] = {kt4.x, kt4.y, kt4.z, kt4.w};
    const float vp[4] = {vp4.x, vp4.y, vp4.z, vp4.w};
    const float s0[4] = {s04.x, s04.y, s04.z, s04.w};
    const float t1[4] = {t14.x, t14.y, t14.z, t14.w};

    float expP[4], A[4], D0[4], D1[4], aC[4], bC[4];
#pragma unroll
    for (int j = 0; j < 4; ++j) {
        float P  = __fdividef(5.0f, t1[j] + EPSF);   // TR/(T1+eps)
        float eP = __expf(-P);
        expP[j]  = eP;
        A[j]     = (1.0f - CF * eP) * s0[j];         // (1-cf*expP)*S0
        float Bd = 1.0f - eP;                        // (1-expP)
        D1[j]    = -CF * Bd;
        D0[j]    = Bd + EPSF;
        aC[j]    = -QS * vp[j];                      // coeff of Cp[t]
        bC[j]    = -QS * (kt[j] * INV60);            // coeff of Ce[t]
    }

    // ---- Time loop: 1 ds_load_b64 + 4x(exp,rcp,3 fma) + 1 b128 store ------
    float* o = out + (size_t)b * NT * HW + pix;
    for (int t = 0; t < NT; ++t) {
        const float2 c = sCpCe[t];                   // broadcast LDS read
        float v[4];
#pragma unroll
        for (int j = 0; j < 4; ++j) {
            float e = expP[j] * __expf(fmaf(aC[j], c.x, bC[j] * c.y)); // exp_PQ
            float n = fmaf(-A[j], e, A[j]);          // (1-exp_PQ)*(1-cf*expP)*S0
            float d = fmaf(D1[j], e, D0[j]);         // (1-cf*exp_PQ)*(1-expP)+eps
            v[j] = __fdividef(n, d);
        }
        float4 r; r.x = v[0]; r.y = v[1]; r.z = v[2]; r.w = v[3];
        *(float4*)o = r;                             // global_store_b128
        o += HW;
    }
}

extern "C" void kernel_launch(void* const* d_in, const int* in_sizes, int n_in,
                              void* d_out, int out_size, void* d_ws, size_t ws_size,
                              hipStream_t stream) {
    const float* params = (const float*)d_in[0];
    const float* Cp     = (const float*)d_in[1];
    const float* S0     = (const float*)d_in[2];
    const float* T1     = (const float*)d_in[3];
    float* out          = (float*)d_out;

    // 4 batches * 65536 px / 4 px-per-thread = 65536 threads
    patlak_st_kernel<<<256, 256, 0, stream>>>(params, Cp, S0, T1, out);
}